// NeighbourAggregation_28054726377693
// MI455X (gfx1250) — compile-verified
//
#include <hip/hip_runtime.h>

typedef __attribute__((ext_vector_type(2))) float v2f;
typedef __attribute__((ext_vector_type(4))) float v4f;
typedef __attribute__((ext_vector_type(8))) float v8f;

// ---------------------------------------------------------------------------
// Phase 1: scatter-add edge features + edge counts with f32 global atomics.
// One thread handles 4 consecutive features of one edge row (float4 NT load).
// ---------------------------------------------------------------------------
__global__ void NeighbourAggregation_scatter_kernel(
    const float* __restrict__ x,         // [E*16] flat
    const int* __restrict__ edge_dst,    // [E]
    float* __restrict__ out,             // [N*16] accumulator (pre-zeroed)
    float* __restrict__ counts,          // [N]     accumulator (pre-zeroed)
    int total)                           // E*16
{
    int t = blockIdx.x * blockDim.x + threadIdx.x;
    int base = t * 4;                    // flat element index, 4 per thread
    if (base >= total) return;

    // 4 consecutive features lie inside one 16-wide row (base is 4-aligned).
    v4f v = __builtin_nontemporal_load((const v4f*)(x + base));
    int e = base >> 4;                   // edge index
    int d = base & 15;                   // feature offset within the row
    int dst = __builtin_nontemporal_load(edge_dst + e);

    float* o = out + dst * 16 + d;
    unsafeAtomicAdd(o + 0, v.x);
    unsafeAtomicAdd(o + 1, v.y);
    unsafeAtomicAdd(o + 2, v.z);
    unsafeAtomicAdd(o + 3, v.w);
    if (d == 0) {
        unsafeAtomicAdd(counts + dst, 1.0f);
    }
}

// ---------------------------------------------------------------------------
// Phase 2: Out = diag(rsqrt(count+1)) x S, one 16x16 node tile per wave,
// via 4 chained V_WMMA_F32_16X16X4_F32 (K = 0,4,8,12), pure f32.
//
// Layouts (ISA 7.12.2, wave32):
//   A 16x4 f32:  lane l -> M = l&15; VGPR j -> K = 2*(l>>4) + j
//   B 4x16 f32:  lane l -> N = l&15; VGPR j -> K = 2*(l>>4) + j
//   C/D 16x16:   lane l, VGPR v -> row M = v + 8*(l>>4), col N = l&15
// ---------------------------------------------------------------------------
__global__ void NeighbourAggregation_finalize_wmma_kernel(
    float* __restrict__ out,             // [N*16] in-place
    const float* __restrict__ counts,    // [N]
    int num_nodes)
{
    int wavesPerBlock = blockDim.x >> 5;
    int gwave = blockIdx.x * wavesPerBlock + (threadIdx.x >> 5);
    int lane  = threadIdx.x & 31;
    int nb    = gwave * 16;              // first node row of this tile
    if (nb >= num_nodes) return;

    int col  = lane & 15;
    int half = lane >> 4;

    if (nb + 16 <= num_nodes) {
        // Full tile: WMMA path. EXEC is all ones for the whole wave here.
        float r = __frsqrt_rn(counts[nb + col] + 1.0f);   // r for row M = col

        v8f c = {0.f, 0.f, 0.f, 0.f, 0.f, 0.f, 0.f, 0.f};
        #pragma unroll
        for (int k0 = 0; k0 < 16; k0 += 4) {
            v2f a, b;
            #pragma unroll
            for (int j = 0; j < 2; ++j) {
                int K = k0 + 2 * half + j;                 // global K column
                a[j] = (col == K) ? r : 0.0f;              // diag(r) slice
                b[j] = out[(nb + K) * 16 + col];           // S rows K, col
            }
            // D = A x B + C   (8 args: neg_a, A, neg_b, B, c_mod, C, reuse_a, reuse_b)
            c = __builtin_amdgcn_wmma_f32_16x16x4_f32(
                    false, a, false, b, (short)0, c, false, false);
        }

        #pragma unroll
        for (int v = 0; v < 8; ++v) {
            int row = v + 8 * half;
            out[(nb + row) * 16 + col] = c[v];
        }
    } else {
        // Partial tail tile (not hit for N=100000): scalar row scaling.
        for (int row = nb + half; row < num_nodes; row += 2) {
            float r = __frsqrt_rn(counts[row] + 1.0f);
            out[row * 16 + col] *= r;
        }
    }
}

// ---------------------------------------------------------------------------
// Launch: zero accumulators, scatter, finalize. All on `stream`, all
// graph-capturable (hipMemsetAsync + kernel launches only).
// ---------------------------------------------------------------------------
extern "C" void kernel_launch(void* const* d_in, const int* in_sizes, int n_in,
                              void* d_out, int out_size, void* d_ws, size_t ws_size,
                              hipStream_t stream) {
    const float* x        = (const float*)d_in[0];
    const int*   edge_dst = (const int*)d_in[1];
    // d_in[2] is num_nodes on device; derive N on host from out_size instead.
    int total     = in_sizes[0];         // E * 16
    int num_nodes = out_size / 16;       // N

    float* out    = (float*)d_out;
    float* counts = (float*)d_ws;        // N floats of scratch

    hipMemsetAsync(d_out, 0, (size_t)out_size * sizeof(float), stream);
    hipMemsetAsync(d_ws, 0, (size_t)num_nodes * sizeof(float), stream);

    int nthreads = total / 4;
    int sblocks  = (nthreads + 255) / 256;
    NeighbourAggregation_scatter_kernel<<<sblocks, 256, 0, stream>>>(
        x, edge_dst, out, counts, total);

    int nwaves  = (num_nodes + 15) / 16; // one 16-node tile per wave
    int fblocks = (nwaves + 7) / 8;      // 8 waves per 256-thread block
    NeighbourAggregation_finalize_wmma_kernel<<<fblocks, 256, 0, stream>>>(
        out, counts, num_nodes);
}